// GIN_31112743092746
// MI455X (gfx1250) — compile-verified
//
#include <hip/hip_runtime.h>

#define N_NODES 200000
#define N_EDGES 3200000
#define FDIM    16
#define NTILES  (N_NODES / 16)   // 12500, exact

typedef float v2f __attribute__((ext_vector_type(2)));
typedef float v8f __attribute__((ext_vector_type(8)));

// ---------------- zero workspace ----------------
__global__ void k_zero(float* __restrict__ p, int n) {
    int i = blockIdx.x * blockDim.x + threadIdx.x;
    if (i < n) p[i] = 0.0f;
}

// ---------------- edge scatter-add: agg[dst] += feat[src] ----------------
// one thread per (edge, feature); lanes 0..15 of each 16-lane group share an
// edge -> contiguous 64B read of the source row, contiguous atomics on dst row.
__global__ __launch_bounds__(256) void k_scatter(const float* __restrict__ feat,
                                                 const int* __restrict__ ei,
                                                 float* __restrict__ agg) {
    unsigned int t = blockIdx.x * 256u + threadIdx.x;
    if (t >= (unsigned int)N_EDGES * FDIM) return;
    int e = (int)(t >> 4);
    int f = (int)(t & 15u);
    int s = ei[e];             // src row of edge_index
    int d = ei[N_EDGES + e];   // dst row of edge_index
    atomicAdd(&agg[(size_t)d * FDIM + f], feat[(size_t)s * FDIM + f]);
}

// ---------------- fused GIN MLP per 16-node tile ----------------
// M = (1+eps)*x + agg ; T = relu(M@wa + ba) ; O = T@wb + bb
// LOGSOFTMAX=false: out = relu(O)   (layer 1)
// LOGSOFTMAX=true : out = log_softmax(O, axis=1)   (layer 2)
template <bool LOGSOFTMAX>
__global__ __launch_bounds__(256) void k_gin_mlp(const float* __restrict__ xin,
                                                 const float* __restrict__ agg,
                                                 const float* __restrict__ eps_p,
                                                 const float* __restrict__ wa,
                                                 const float* __restrict__ ba,
                                                 const float* __restrict__ wb,
                                                 const float* __restrict__ bb,
                                                 float* __restrict__ out,
                                                 int n_tiles) {
    // per-wave 16x16 staging tile, padded to 17 to avoid bank conflicts
    __shared__ float lds[8][16][17];

    const int wid  = threadIdx.x >> 5;
    const int lane = threadIdx.x & 31;
    const int tile = blockIdx.x * 8 + wid;
    if (tile >= n_tiles) return;          // whole-wave uniform; wave terminates

    const int col = lane & 15;            // A-matrix row (M) and B/C/D column (N)
    const int hi  = lane >> 4;            // half-wave selector
    const float epsp = 1.0f + eps_p[0];

    const float* xrow = xin + (size_t)(tile * 16 + col) * FDIM;
    const float* grow = agg + (size_t)(tile * 16 + col) * FDIM;

    // ---- GEMM 1: (1+eps)x + agg  times  wa  (K=16 as 4 chained K=4 WMMAs) ----
    v8f acc = {};
#pragma unroll
    for (int c = 0; c < 4; ++c) {
        const int kb = c * 4 + hi * 2;    // ISA 32-bit A 16x4 layout: lanes>=16 hold K+2
        v2f xv = *(const v2f*)(xrow + kb);
        v2f gv = *(const v2f*)(grow + kb);
        v2f a;
        a.x = epsp * xv.x + gv.x;
        a.y = epsp * xv.y + gv.y;
        v2f b;                            // B 4x16: V0 = rows kb/kb+2, V1 = rows kb+1/kb+3
        b.x = wa[(kb + 0) * 16 + col];
        b.y = wa[(kb + 1) * 16 + col];
        acc = __builtin_amdgcn_wmma_f32_16x16x4_f32(false, a, false, b,
                                                    (short)0, acc, false, false);
    }

    // bias + ReLU, then C/D layout -> row-major LDS (VGPR v holds rows v and v+8)
    const float biasA = ba[col];
#pragma unroll
    for (int v = 0; v < 8; ++v) {
        float t = acc[v] + biasA;
        lds[wid][v + hi * 8][col] = t > 0.0f ? t : 0.0f;
    }
    __syncthreads();

    // ---- GEMM 2: T times wb ----
    v8f acc2 = {};
#pragma unroll
    for (int c = 0; c < 4; ++c) {
        const int kb = c * 4 + hi * 2;
        v2f a;
        a.x = lds[wid][col][kb + 0];
        a.y = lds[wid][col][kb + 1];
        v2f b;
        b.x = wb[(kb + 0) * 16 + col];
        b.y = wb[(kb + 1) * 16 + col];
        acc2 = __builtin_amdgcn_wmma_f32_16x16x4_f32(false, a, false, b,
                                                     (short)0, acc2, false, false);
    }

    const float biasB = bb[col];
    float* orow = out + (size_t)tile * 16 * FDIM;

    if (LOGSOFTMAX) {
        // each acc2[v] is one row spread across the 16 lanes of this half-wave
#pragma unroll
        for (int v = 0; v < 8; ++v) {
            float val = acc2[v] + biasB;
            float m = val;
#pragma unroll
            for (int off = 1; off < 16; off <<= 1)
                m = fmaxf(m, __shfl_xor(m, off, 16));
            float s = __expf(val - m);
#pragma unroll
            for (int off = 1; off < 16; off <<= 1)
                s += __shfl_xor(s, off, 16);
            orow[(v + hi * 8) * FDIM + col] = val - m - __logf(s);
        }
    } else {
#pragma unroll
        for (int v = 0; v < 8; ++v) {
            float t = acc2[v] + biasB;
            orow[(v + hi * 8) * FDIM + col] = t > 0.0f ? t : 0.0f;
        }
    }
}

// ---------------- launcher ----------------
extern "C" void kernel_launch(void* const* d_in, const int* in_sizes, int n_in,
                              void* d_out, int out_size, void* d_ws, size_t ws_size,
                              hipStream_t stream) {
    const float* x    = (const float*)d_in[0];
    const int*   ei   = (const int*)d_in[1];
    const float* eps1 = (const float*)d_in[2];
    const float* w1a  = (const float*)d_in[3];
    const float* b1a  = (const float*)d_in[4];
    const float* w1b  = (const float*)d_in[5];
    const float* b1b  = (const float*)d_in[6];
    const float* eps2 = (const float*)d_in[7];
    const float* w2a  = (const float*)d_in[8];
    const float* b2a  = (const float*)d_in[9];
    const float* w2b  = (const float*)d_in[10];
    const float* b2b  = (const float*)d_in[11];
    float* out = (float*)d_out;

    // workspace: [agg: N*F floats][h1: N*F floats]  (agg reused for both layers)
    float* agg = (float*)d_ws;
    float* h1  = agg + (size_t)N_NODES * FDIM;

    const int nfeat       = N_NODES * FDIM;                     // 3.2M
    const int zero_blocks = (nfeat + 255) / 256;
    const unsigned int scat_threads = (unsigned int)N_EDGES * FDIM;  // 51.2M
    const int scat_blocks = (int)((scat_threads + 255u) / 256u);
    const int mlp_blocks  = (NTILES + 7) / 8;

    // ---- layer 1 ----
    k_zero<<<zero_blocks, 256, 0, stream>>>(agg, nfeat);
    k_scatter<<<scat_blocks, 256, 0, stream>>>(x, ei, agg);
    k_gin_mlp<false><<<mlp_blocks, 256, 0, stream>>>(x, agg, eps1, w1a, b1a, w1b, b1b,
                                                     h1, NTILES);
    // ---- layer 2 ----
    k_zero<<<zero_blocks, 256, 0, stream>>>(agg, nfeat);
    k_scatter<<<scat_blocks, 256, 0, stream>>>(h1, ei, agg);
    k_gin_mlp<true><<<mlp_blocks, 256, 0, stream>>>(h1, agg, eps2, w2a, b2a, w2b, b2b,
                                                    out, NTILES);
}